// MDN2Layer_25039659336010
// MI455X (gfx1250) — compile-verified
//
#include <hip/hip_runtime.h>
#include <math.h>

// ---------------------------------------------------------------------------
// MDN 2-layer: y = x@W + b  (16x128 @ 128x96), then per-(b,k) Gaussian params
// + softmax weights, then mixture pdf evaluated on a 302x394 grid per batch.
//
// Kernel 1 (setup): six wave32 waves, each computes one 16x16 tile of y via
//   V_WMMA_F32_16X16X4_F32 (K=128 -> 32 wmma steps). Bias fused. Stage 2
//   computes softmax + folded coefficients, writes params to workspace.
// Kernel 2 (eval): one thread per (b, grid point); 16-mixture loop with
//   params broadcast from LDS; v_exp_f32 via __expf.
// ---------------------------------------------------------------------------

typedef float v2f __attribute__((ext_vector_type(2)));
typedef float v8f __attribute__((ext_vector_type(8)));

#define NXg 302
#define NYg 394
#define G_TOTAL (NXg * NYg)        // 118988
#define B_SZ 16
#define D_SZ 128
#define KMIX 16
#define UNITS 96
#define INV_TWO_PI 0.15915494309189535f

__global__ __launch_bounds__(192)
void mdn_setup_kernel(const float* __restrict__ inp,   // [16][128]
                      const float* __restrict__ w,     // [128][96]
                      const float* __restrict__ bias,  // [96]
                      float* __restrict__ params)      // [16][16][8]
{
    __shared__ float y_lds[B_SZ * UNITS];

    const unsigned tid  = threadIdx.x;
    const unsigned wave = tid >> 5;          // 0..5 : N-tile index
    const unsigned lane = tid & 31u;
    const unsigned half = lane >> 4;         // 0 -> K pair {0,1}, 1 -> {2,3}
    const unsigned mn   = lane & 15u;        // M row (A) / N col (B, C/D)
    const unsigned n_g  = wave * 16u + mn;   // global output column 0..95

    // --- WMMA FP32 GEMM: D(16x16) += A(16x4) * B(4x16), 32 K-steps ---------
    v8f acc = {};
    for (unsigned kk = 0; kk < D_SZ; kk += 4u) {
        const unsigned k0 = kk + 2u * half;
        v2f a, b;
        // A 16x4 f32 layout: lane<16 holds K=0,1 ; lane>=16 holds K=2,3
        a.x = inp[mn * D_SZ + k0 + 0];
        a.y = inp[mn * D_SZ + k0 + 1];
        // B 4x16 f32 layout (mirror of A with N across lanes)
        b.x = w[(k0 + 0) * UNITS + n_g];
        b.y = w[(k0 + 1) * UNITS + n_g];
        // 8 args: (neg_a, A, neg_b, B, c_mod, C, reuse_a, reuse_b)
        acc = __builtin_amdgcn_wmma_f32_16x16x4_f32(false, a, false, b,
                                                    (short)0, acc,
                                                    false, false);
    }

    // C/D layout: VGPR r -> M = r + 8*half ; N = lane&15
    const float bn = bias[n_g];
#pragma unroll
    for (int r = 0; r < 8; ++r) {
        const unsigned m = (unsigned)r + 8u * half;
        y_lds[m * UNITS + n_g] = acc[r] + bn;
    }
    __syncthreads();

    // --- Stage 2: per-batch softmax + folded Gaussian params ----------------
    if (tid < B_SZ) {
        const unsigned bb = tid;
        const float* yb = &y_lds[bb * UNITS];

        // softmax over pi logits yb[80..95]
        float mx = -INFINITY;
#pragma unroll
        for (int k = 0; k < KMIX; ++k) mx = fmaxf(mx, yb[80 + k]);
        float e[KMIX];
        float s = 0.0f;
#pragma unroll
        for (int k = 0; k < KMIX; ++k) { e[k] = __expf(yb[80 + k] - mx); s += e[k]; }
        const float inv_s = 1.0f / s;

#pragma unroll
        for (int k = 0; k < KMIX; ++k) {
            const float mux  = yb[2 * k + 0];
            const float muy  = yb[2 * k + 1];
            const float a_   = yb[32 + 3 * k + 0];
            const float iL22 = __expf(-yb[32 + 3 * k + 1]);   // 1/L22
            const float iL11 = __expf(-yb[32 + 3 * k + 2]);   // 1/L11
            const float wk   = e[k] * inv_s;
            const float coeff = wk * iL11 * iL22 * INV_TWO_PI; // w/(2*pi*L11*L22)
            float* p = &params[(bb * KMIX + k) * 8];
            p[0] = mux;  p[1] = muy;  p[2] = a_;
            p[3] = iL11; p[4] = iL22; p[5] = coeff;
            p[6] = 0.0f; p[7] = 0.0f;
        }
    }
}

__global__ __launch_bounds__(256)
void mdn_eval_kernel(const float* __restrict__ params, // [16][16][8]
                     float* __restrict__ out)          // [16][NX][NY]
{
    __shared__ float p_lds[KMIX * 8];
    const unsigned bb = blockIdx.y;
    if (threadIdx.x < KMIX * 8)
        p_lds[threadIdx.x] = params[bb * (KMIX * 8) + threadIdx.x];
    __syncthreads();

    const unsigned g = blockIdx.x * blockDim.x + threadIdx.x;
    if (g >= G_TOTAL) return;

    const unsigned ix = g / NYg;
    const unsigned iy = g - ix * NYg;
    const float gx = (float)ix * (1.0f / (float)(NXg - 1));
    const float gy = (float)iy * (1.0f / (float)(NYg - 1));

    float acc = 0.0f;
#pragma unroll
    for (int k = 0; k < KMIX; ++k) {
        const float* p = &p_lds[k * 8];          // LDS broadcast (uniform addr)
        const float z1 = (gx - p[0]) * p[3];
        const float z2 = ((gy - p[1]) - p[2] * z1) * p[4];
        acc = fmaf(p[5], __expf(-0.5f * (z1 * z1 + z2 * z2)), acc);
    }
    out[bb * (unsigned)G_TOTAL + g] = acc;
}

extern "C" void kernel_launch(void* const* d_in, const int* in_sizes, int n_in,
                              void* d_out, int out_size, void* d_ws, size_t ws_size,
                              hipStream_t stream) {
    const float* inp  = (const float*)d_in[0];   // (16,128) f32
    const float* w    = (const float*)d_in[1];   // (128,96) f32
    const float* bias = (const float*)d_in[2];   // (96,)    f32
    float* params = (float*)d_ws;                // 16*16*8 floats = 8 KB
    float* out    = (float*)d_out;               // 16*302*394 floats

    mdn_setup_kernel<<<1, 192, 0, stream>>>(inp, w, bias, params);

    dim3 grid((G_TOTAL + 255) / 256, B_SZ, 1);
    mdn_eval_kernel<<<grid, 256, 0, stream>>>(params, out);
}